// GCN_60455959658660
// MI455X (gfx1250) — compile-verified
//
#include <hip/hip_runtime.h>

#define NN 100000     // nodes
#define NE 600000     // edges
#define HID 128       // hidden size (== num tasks)
#define NG 4000       // graphs
#define HV (HID / 4)  // float4 columns per row

typedef __attribute__((ext_vector_type(2))) float v2f;
typedef __attribute__((ext_vector_type(8))) float v8f;

// ---------------------------------------------------------------- utilities
__global__ void zero_f32(float* __restrict__ p, int n) {
  int i = blockIdx.x * blockDim.x + threadIdx.x;
  if (i < n) p[i] = 0.0f;
}

__global__ void deg_accum(const int* __restrict__ dst, float* __restrict__ deg) {
  int e = blockIdx.x * blockDim.x + threadIdx.x;
  if (e < NE) atomicAdd(&deg[dst[e]], 1.0f);
}

__global__ void deg_to_dinv(float* __restrict__ d) {
  int n = blockIdx.x * blockDim.x + threadIdx.x;
  if (n < NN) d[n] = rsqrtf(d[n] + 1.0f);
}

// ------------------------------------------------- AtomEncoder: sum of 9 embeddings
__global__ void atom_encoder(const int* __restrict__ x, const float* __restrict__ emb,
                             float* __restrict__ h) {
  const int offs[9] = {0, 119, 124, 136, 148, 158, 164, 170, 172};
  int id = blockIdx.x * blockDim.x + threadIdx.x;   // over NN*HV float4 slots
  if (id >= NN * HV) return;
  int n = id >> 5;
  int c = id & (HV - 1);
  float4 acc = make_float4(0.f, 0.f, 0.f, 0.f);
#pragma unroll
  for (int f = 0; f < 9; ++f) {
    int row = x[n * 9 + f] + offs[f];
    float4 e = ((const float4*)emb)[row * HV + c];
    acc.x += e.x; acc.y += e.y; acc.z += e.z; acc.w += e.w;
  }
  ((float4*)h)[id] = acc;
}

// ------------------------------------------------- fp32 WMMA GEMM: C[M x 128] = A[M x 128] * B[128 x 128] (+ bias)
// block = 256 threads = 8 waves; wave w computes output columns [16w, 16w+16).
// B (64 KB) is staged once per block in LDS; block grid-strides over 16-row A strips.
__global__ __launch_bounds__(256) void gemm_wmma(const float* __restrict__ A,
                                                 const float* __restrict__ B,
                                                 const float* __restrict__ bias,
                                                 float* __restrict__ C, int M) {
  __shared__ float Bs[HID][HID];        // 64 KB
  __shared__ float As[16][HID + 4];     // padded: avoid 16-way bank conflict on column reads
  const int tid  = threadIdx.x;
  const int lane = tid & 31;
  const int wave = tid >> 5;            // 0..7 -> output column block
  const int half = lane >> 4;           // K-half select per ISA f32 WMMA layout
  const int l    = lane & 15;
  const int n    = wave * 16 + l;       // this lane's output column

  // stage B (row-major [K][N]) into LDS
  for (int i = tid; i < HID * HV; i += 256) {
    float4 v = ((const float4*)B)[i];
    int r = i >> 5, c4 = i & (HV - 1);
    *(float4*)&Bs[r][c4 * 4] = v;
  }
  __syncthreads();

  const int strips = M >> 4;
  for (int s = blockIdx.x; s < strips; s += gridDim.x) {
    // stage 16x128 A strip
    for (int i = tid; i < 16 * HV; i += 256) {
      float4 v = ((const float4*)A)[(size_t)s * (16 * HV) + i];
      int r = i >> 5, c4 = i & (HV - 1);
      *(float4*)&As[r][c4 * 4] = v;
    }
    __syncthreads();

    v8f acc = {};
#pragma unroll
    for (int k = 0; k < HID; k += 4) {
      // A 16x4 fragment: lanes 0-15 -> K = k,k+1 ; lanes 16-31 -> K = k+2,k+3 ; M = lane%16
      v2f a, b;
      a.x = As[l][k + 2 * half];
      a.y = As[l][k + 2 * half + 1];
      // B 4x16 fragment: lanes 0-15 -> rows k,k+1 ; lanes 16-31 -> rows k+2,k+3 ; N = lane%16 (+wave tile)
      b.x = Bs[k + 2 * half][n];
      b.y = Bs[k + 2 * half + 1][n];
      acc = __builtin_amdgcn_wmma_f32_16x16x4_f32(false, a, false, b, (short)0, acc,
                                                  false, false);
    }

    float bv = (bias != nullptr) ? bias[n] : 0.0f;
    // C/D layout: VGPR v holds row v + 8*(lane>=16), col = lane%16
#pragma unroll
    for (int v = 0; v < 8; ++v) {
      int row = (s << 4) + v + 8 * half;
      C[(size_t)row * HID + n] = acc[v] + bv;
    }
    __syncthreads();
  }
}

// ------------------------------------------------- h = dinv^2 * t + b  (self-loop + bias; re-inits h)
__global__ void self_bias(const float* __restrict__ t, const float* __restrict__ dinv,
                          const float* __restrict__ b, float* __restrict__ h) {
  int id = blockIdx.x * blockDim.x + threadIdx.x;
  if (id >= NN * HV) return;
  int n = id >> 5, c4 = id & (HV - 1);
  float di = dinv[n];
  float w = di * di;
  float4 tv = ((const float4*)t)[id];
  float4 bv = ((const float4*)b)[c4];
  float4 o = make_float4(fmaf(tv.x, w, bv.x), fmaf(tv.y, w, bv.y),
                         fmaf(tv.z, w, bv.z), fmaf(tv.w, w, bv.w));
  ((float4*)h)[id] = o;
}

// ------------------------------------------------- edge scatter: h[dst] += t[src] * dinv[src]*dinv[dst]
// 8 edges per 256-thread block; 32 lanes per edge, float4 per lane, f32 global atomics (hit L2).
__global__ __launch_bounds__(256) void edge_scatter(const float* __restrict__ t,
                                                    const float* __restrict__ dinv,
                                                    const int* __restrict__ src,
                                                    const int* __restrict__ dst,
                                                    float* __restrict__ h) {
  int e = blockIdx.x * 8 + (threadIdx.x >> 5);
  int lane = threadIdx.x & 31;
  if (e >= NE) return;
  int s = src[e], d = dst[e];
  float nrm = dinv[s] * dinv[d];
  float4 m = ((const float4*)t)[(size_t)s * HV + lane];
  float* out = h + (size_t)d * HID + lane * 4;
  atomicAdd(out + 0, m.x * nrm);
  atomicAdd(out + 1, m.y * nrm);
  atomicAdd(out + 2, m.z * nrm);
  atomicAdd(out + 3, m.w * nrm);
}

__global__ void relu_ip(float* __restrict__ h, int n4) {
  int id = blockIdx.x * blockDim.x + threadIdx.x;
  if (id >= n4) return;
  float4 v = ((float4*)h)[id];
  v.x = fmaxf(v.x, 0.f); v.y = fmaxf(v.y, 0.f);
  v.z = fmaxf(v.z, 0.f); v.w = fmaxf(v.w, 0.f);
  ((float4*)h)[id] = v;
}

// ------------------------------------------------- global mean pool
__global__ void graph_count(const int* __restrict__ batch, float* __restrict__ cnt) {
  int n = blockIdx.x * blockDim.x + threadIdx.x;
  if (n < NN) atomicAdd(&cnt[batch[n]], 1.0f);
}

__global__ void graph_pool(const float* __restrict__ h, const int* __restrict__ batch,
                           float* __restrict__ pooled) {
  int id = blockIdx.x * blockDim.x + threadIdx.x;
  if (id >= NN * HV) return;
  int n = id >> 5, c4 = id & (HV - 1);
  int g = batch[n];
  float4 v = ((const float4*)h)[id];
  float* out = pooled + (size_t)g * HID + c4 * 4;
  atomicAdd(out + 0, v.x);
  atomicAdd(out + 1, v.y);
  atomicAdd(out + 2, v.z);
  atomicAdd(out + 3, v.w);
}

__global__ void pool_div(float* __restrict__ pooled, const float* __restrict__ cnt) {
  int id = blockIdx.x * blockDim.x + threadIdx.x;
  if (id >= NG * HV) return;
  int g = id >> 5;
  float inv = 1.0f / fmaxf(cnt[g], 1.0f);
  float4 v = ((float4*)pooled)[id];
  v.x *= inv; v.y *= inv; v.z *= inv; v.w *= inv;
  ((float4*)pooled)[id] = v;
}

// ---------------------------------------------------------------- launcher
extern "C" void kernel_launch(void* const* d_in, const int* in_sizes, int n_in,
                              void* d_out, int out_size, void* d_ws, size_t ws_size,
                              hipStream_t stream) {
  const int*   x     = (const int*)d_in[0];
  const int*   ei    = (const int*)d_in[1];
  const int*   batch = (const int*)d_in[2];
  const float* emb   = (const float*)d_in[3];
  const float* W1    = (const float*)d_in[4];
  const float* b1    = (const float*)d_in[5];
  const float* W2    = (const float*)d_in[6];
  const float* b2    = (const float*)d_in[7];
  const float* W3    = (const float*)d_in[8];
  const float* b3    = (const float*)d_in[9];
  const float* Wl    = (const float*)d_in[10];
  const float* bl    = (const float*)d_in[11];
  float* out = (float*)d_out;

  // workspace layout (floats): dinv[NN] | h[NN*HID] | t[NN*HID] | pooled[NG*HID] | cnt[NG]
  float* dinv   = (float*)d_ws;
  float* h      = dinv + NN;
  float* t      = h + (size_t)NN * HID;
  float* pooled = t + (size_t)NN * HID;
  float* cnt    = pooled + (size_t)NG * HID;

  const int* src = ei;          // edge_index[0]
  const int* dstp = ei + NE;    // edge_index[1]

  dim3 b256(256);

  // degrees -> dinv (in place)
  zero_f32<<<(NN + 255) / 256, b256, 0, stream>>>(dinv, NN);
  deg_accum<<<(NE + 255) / 256, b256, 0, stream>>>(dstp, dinv);
  deg_to_dinv<<<(NN + 255) / 256, b256, 0, stream>>>(dinv);

  // atom encoder -> h
  atom_encoder<<<(NN * HV + 255) / 256, b256, 0, stream>>>(x, emb, h);

  // three GCN conv layers
  const float* Ws[3] = {W1, W2, W3};
  const float* bs[3] = {b1, b2, b3};
  for (int layer = 0; layer < 3; ++layer) {
    gemm_wmma<<<1024, b256, 0, stream>>>(h, Ws[layer], nullptr, t, NN);
    self_bias<<<(NN * HV + 255) / 256, b256, 0, stream>>>(t, dinv, bs[layer], h);
    edge_scatter<<<(NE + 7) / 8, b256, 0, stream>>>(t, dinv, src, dstp, h);
    if (layer < 2) relu_ip<<<(NN * HV + 255) / 256, b256, 0, stream>>>(h, NN * HV);
  }

  // global mean pool (pooled and cnt are contiguous -> single zero pass)
  zero_f32<<<(NG * HID + NG + 255) / 256, b256, 0, stream>>>(pooled, NG * HID + NG);
  graph_count<<<(NN + 255) / 256, b256, 0, stream>>>(batch, cnt);
  graph_pool<<<(NN * HV + 255) / 256, b256, 0, stream>>>(h, batch, pooled);
  pool_div<<<(NG * HV + 255) / 256, b256, 0, stream>>>(pooled, cnt);

  // final linear: out[G x 128] = pooled @ Wl + bl
  gemm_wmma<<<250, b256, 0, stream>>>(pooled, Wl, bl, out, NG);
}